// VolumeRenderer_84507776516622
// MI455X (gfx1250) — compile-verified
//
#include <hip/hip_runtime.h>
#include <cstdint>
#include <cstddef>

#ifndef __has_builtin
#define __has_builtin(x) 0
#endif

#define HAVE_ASYNC_LDS __has_builtin(__builtin_amdgcn_global_load_async_to_lds_b128)
#define HAVE_WAIT_ASYNC __has_builtin(__builtin_amdgcn_s_wait_asynccnt)

typedef float v4f __attribute__((ext_vector_type(4)));
// Exact pointee type the async-LDS builtin expects (per hipcc diagnostic).
typedef int v4i __attribute__((vector_size(16)));

namespace {

constexpr float kNear = 2.0f;
constexpr float kFar  = 6.0f;
constexpr int   kZ    = 64;

constexpr int kThreads    = 64;                   // 2 wave32 per block, 1 ray/thread
constexpr int kChunkZ     = 8;                    // z-samples per chunk
constexpr int kNChunks    = kZ / kChunkZ;         // 8
constexpr int kRayRadF    = kZ * 3;               // 192 floats of radiance per ray
constexpr int kRayRadB    = kRayRadF * 4;         // 768 B of radiance per ray
constexpr int kChunkRadB  = kChunkZ * 3 * 4;      // 96 B per ray per chunk
constexpr int kLdsRowB    = kRayRadB + 16;        // 784 B padded row -> conflict-free b128 reads
constexpr int kWaveBufB   = 32 * kLdsRowB;        // 25088 B per wave
constexpr int kSegsPerLane = 6;                   // 6 x 16B async copies / lane / chunk

// exp(-sigma*dist) = exp2(sigma * (-dist*log2(e))) : one v_mul + one v_exp_f32
constexpr float kLog2e   = 1.4426950408889634f;
constexpr float kDzMid   = (kFar - kNear) / (float)(kZ - 1);     // 4/63
constexpr float kCexpMid = -kDzMid * kLog2e;
constexpr float kCexpLast = -1e10f * kLog2e;

__device__ __forceinline__ float fast_exp2(float x) {
#if __has_builtin(__builtin_amdgcn_exp2f)
  return __builtin_amdgcn_exp2f(x);   // raw v_exp_f32
#else
  return exp2f(x);
#endif
}

template <int N>
__device__ __forceinline__ void wait_asynccnt() {
#if HAVE_WAIT_ASYNC
  __builtin_amdgcn_s_wait_asynccnt(N);
#elif HAVE_ASYNC_LDS
  asm volatile("s_wait_asynccnt %0" :: "n"(N) : "memory");
#endif
}

#if HAVE_ASYNC_LDS
// Issue the 6 copies of chunk C.  The per-lane base addresses are fixed; the
// chunk advance (C*96B) rides in the instruction's IOFFSET, which hardware
// applies to BOTH the global and the LDS address -> zero per-chunk VALU.
template <int C>
__device__ __forceinline__ void issue_chunk(const char* const g[kSegsPerLane],
                                            const uint32_t l[kSegsPerLane]) {
  #pragma unroll
  for (int k = 0; k < kSegsPerLane; ++k) {
    __builtin_amdgcn_global_load_async_to_lds_b128(
        (__attribute__((address_space(1))) v4i*)(uintptr_t)g[k],
        (__attribute__((address_space(3))) v4i*)l[k],
        C * kChunkRadB, 0);
  }
}

// Stage the wave's full 32-ray radiance tile (48 x b128 per lane, in order).
// Lane remap u = lane + 32k -> (ray u/6, 16B segment u%6): adjacent lanes fetch
// adjacent 16B of memory (96B runs, coalesced) into padded 784B LDS rows.
__device__ __forceinline__ void stage_all_async(const char* rad_wave, char* wave_lds, int lane) {
  const char* g[kSegsPerLane];
  uint32_t    l[kSegsPerLane];
  #pragma unroll
  for (int k = 0; k < kSegsPerLane; ++k) {
    const int u = lane + 32 * k;
    const int r = u / 6;
    const int p = u % 6;
    g[k] = rad_wave + (size_t)r * kRayRadB + p * 16;
    l[k] = (uint32_t)(uintptr_t)(wave_lds + r * kLdsRowB + p * 16);
  }
  issue_chunk<0>(g, l); issue_chunk<1>(g, l); issue_chunk<2>(g, l); issue_chunk<3>(g, l);
  issue_chunk<4>(g, l); issue_chunk<5>(g, l); issue_chunk<6>(g, l); issue_chunk<7>(g, l);
}
#endif

template <int C>
__device__ __forceinline__ void process_chunk(const float* __restrict__ sig_ray,
                                              const char* lane_lds,
                                              const float* __restrict__ rad_ray,
                                              float& T, float& ar, float& ag,
                                              float& ab, float& ad) {
  // Chunks complete in issue order: chunks <= C are done once <= 6*(7-C) remain.
  wait_asynccnt<kSegsPerLane * (kNChunks - 1 - C)>();

  v4f s0 = __builtin_nontemporal_load((const v4f*)(sig_ray + C * kChunkZ));
  v4f s1 = __builtin_nontemporal_load((const v4f*)(sig_ray + C * kChunkZ + 4));

  float rr[kChunkZ * 3];
#if HAVE_ASYNC_LDS
  (void)rad_ray;
  const v4f* lp = (const v4f*)(lane_lds + C * kChunkRadB);   // 6 x ds_load_b128
  #pragma unroll
  for (int k = 0; k < 6; ++k) {
    v4f t = lp[k];
    rr[4 * k + 0] = t.x; rr[4 * k + 1] = t.y;
    rr[4 * k + 2] = t.z; rr[4 * k + 3] = t.w;
  }
#else
  (void)lane_lds;
  const v4f* gp = (const v4f*)(rad_ray + C * kChunkZ * 3);
  #pragma unroll
  for (int k = 0; k < 6; ++k) {
    v4f t = __builtin_nontemporal_load(gp + k);
    rr[4 * k + 0] = t.x; rr[4 * k + 1] = t.y;
    rr[4 * k + 2] = t.z; rr[4 * k + 3] = t.w;
  }
#endif
  #pragma unroll
  for (int j = 0; j < kChunkZ; ++j) {
    const int z = C * kChunkZ + j;
    const float sg    = (j < 4) ? s0[j] : s1[j - 4];
    const float cexp  = (z == kZ - 1) ? kCexpLast : kCexpMid;
    const float e     = fast_exp2(sg * cexp);         // exp(-sg*dist)
    const float alpha = 1.0f - e;
    T *= (1.0f - alpha + 1e-10f);                     // inclusive cumprod (matches reference)
    const float w = alpha * T;
    ar += w * rr[3 * j + 0];
    ag += w * rr[3 * j + 1];
    ab += w * rr[3 * j + 2];
    ad += w * (kNear + (float)z * kDzMid);
  }
}

}  // namespace

__global__ __launch_bounds__(kThreads) void volume_render_kernel(
    const float* __restrict__ sigmas,      // [nRays, 64]
    const float* __restrict__ radiances,   // [nRays, 64, 3]
    float* __restrict__ out_rgb,           // [nRays, 3]
    float* __restrict__ out_depth,         // [nRays]
    int n_rays)
{
  __shared__ __align__(16) char smem[(kThreads / 32) * kWaveBufB];  // 49 KB

  const int lane     = threadIdx.x & 31;
  const int wave     = threadIdx.x >> 5;
  const int ray_base = blockIdx.x * kThreads;
  const int ray      = ray_base + (int)threadIdx.x;

  float T = 1.0f, ar = 0.0f, ag = 0.0f, ab = 0.0f, ad = 0.0f;

  const bool full_block = (ray_base + kThreads) <= n_rays;

  if (full_block) {
    char* wave_lds = smem + wave * kWaveBufB;
    const char* rad_wave = (const char*)(radiances + (size_t)(ray_base + wave * 32) * kRayRadF);
    const float* sig_ray = sigmas + (size_t)ray * kZ;
    const float* rad_ray = radiances + (size_t)ray * kRayRadF;
    const char* lane_lds = wave_lds + lane * kLdsRowB;
    (void)rad_wave;

#if HAVE_ASYNC_LDS
    stage_all_async(rad_wave, wave_lds, lane);   // 48 in-flight b128 copies / lane
#endif
    process_chunk<0>(sig_ray, lane_lds, rad_ray, T, ar, ag, ab, ad);
    process_chunk<1>(sig_ray, lane_lds, rad_ray, T, ar, ag, ab, ad);
    process_chunk<2>(sig_ray, lane_lds, rad_ray, T, ar, ag, ab, ad);
    process_chunk<3>(sig_ray, lane_lds, rad_ray, T, ar, ag, ab, ad);
    process_chunk<4>(sig_ray, lane_lds, rad_ray, T, ar, ag, ab, ad);
    process_chunk<5>(sig_ray, lane_lds, rad_ray, T, ar, ag, ab, ad);
    process_chunk<6>(sig_ray, lane_lds, rad_ray, T, ar, ag, ab, ad);
    process_chunk<7>(sig_ray, lane_lds, rad_ray, T, ar, ag, ab, ad);
  } else {
    // Tail path (unused for the reference sizes: 262144 % 64 == 0)
    if (ray >= n_rays) return;
    const float* sig_ray = sigmas + (size_t)ray * kZ;
    const float* rad_ray = radiances + (size_t)ray * kRayRadF;
    for (int z = 0; z < kZ; ++z) {
      const float cexp  = (z == kZ - 1) ? kCexpLast : kCexpMid;
      const float alpha = 1.0f - fast_exp2(sig_ray[z] * cexp);
      T *= (1.0f - alpha + 1e-10f);
      const float w = alpha * T;
      ar += w * rad_ray[3 * z + 0];
      ag += w * rad_ray[3 * z + 1];
      ab += w * rad_ray[3 * z + 2];
      ad += w * (kNear + (float)z * kDzMid);
    }
  }

  out_rgb[(size_t)ray * 3 + 0] = ar;
  out_rgb[(size_t)ray * 3 + 1] = ag;
  out_rgb[(size_t)ray * 3 + 2] = ab;
  out_depth[ray] = ad;
}

extern "C" void kernel_launch(void* const* d_in, const int* in_sizes, int n_in,
                              void* d_out, int out_size, void* d_ws, size_t ws_size,
                              hipStream_t stream) {
  (void)n_in; (void)out_size; (void)d_ws; (void)ws_size;
  const float* sigmas    = (const float*)d_in[0];   // [B,R,64,1] f32
  const float* radiances = (const float*)d_in[1];   // [B,R,64,3] f32
  float* out = (float*)d_out;                       // rgb [B,R,3] ++ depth [B,R,1]

  const int n_rays = in_sizes[0] / kZ;              // B*R = 262144
  float* out_rgb   = out;
  float* out_depth = out + (size_t)n_rays * 3;

  dim3 block(kThreads);
  dim3 grid((n_rays + kThreads - 1) / kThreads);
  hipLaunchKernelGGL(volume_render_kernel, grid, block, 0, stream,
                     sigmas, radiances, out_rgb, out_depth, n_rays);
}